// GruAeModel_81767587381816
// MI455X (gfx1250) — compile-verified
//
#include <hip/hip_runtime.h>
#include <hip/hip_bf16.h>
#include <cstddef>

// Problem dims (fixed by the reference)
#define B_N 1024
#define S_N 1024
#define R_N 12
#define F_N 36   // input feature width = 12 + 4 + 4 + 8 + 8; F = 9 K-chunks of 4
#define H_N 20   // hidden;  H = 5 K-chunks of 4
#define G_N 60   // 3*H gates, padded to 4 N-tiles of 16

typedef __attribute__((ext_vector_type(2))) float v2f;
typedef __attribute__((ext_vector_type(8))) float v8f;

// f32 WMMA, D(16x16) += A(16x4) * B(4x16), full fp32 (precision-safe for the
// 1024-step recurrence). 8-arg VOP3P form per CDNA5_HIP.md.
__device__ __forceinline__ v8f wmma4(v2f a, v2f b, v8f c) {
  return __builtin_amdgcn_wmma_f32_16x16x4_f32(
      /*neg_a=*/false, a, /*neg_b=*/false, b,
      /*c_mod=*/(short)0, c, /*reuse_a=*/false, /*reuse_b=*/false);
}

__device__ __forceinline__ float sigm(float v) { return 1.0f / (1.0f + __expf(-v)); }

// ---------------------------------------------------------------------------
// Kernel 1: embedding lookup + concat -> origin (B,S,F). Pure bandwidth.
// ---------------------------------------------------------------------------
__global__ void embed_concat_kernel(const float* __restrict__ x,
                                    const int*  __restrict__ oh,
                                    const float* __restrict__ e0,
                                    const float* __restrict__ e1,
                                    const float* __restrict__ e2,
                                    const float* __restrict__ e3,
                                    float* __restrict__ origin) {
  int idx = blockIdx.x * blockDim.x + threadIdx.x;   // (b*S + s)
  if (idx >= B_N * S_N) return;
  const float4* xr = (const float4*)(x + (size_t)idx * R_N);   // 48B rows, 16B aligned
  float4* orow = (float4*)(origin + (size_t)idx * F_N);        // 144B rows, 16B aligned
  orow[0] = xr[0];
  orow[1] = xr[1];
  orow[2] = xr[2];
  const int* o = oh + (size_t)idx * 4;
  orow[3] = *(const float4*)(e0 + o[0] * 4);
  orow[4] = *(const float4*)(e1 + o[1] * 4);
  const float4* p2 = (const float4*)(e2 + o[2] * 8);
  const float4* p3 = (const float4*)(e3 + o[3] * 8);
  orow[5] = p2[0];
  orow[6] = p2[1];
  orow[7] = p3[0];
  orow[8] = p3[1];
}

// ---------------------------------------------------------------------------
// Per-lane WMMA layout helpers (wave32):
//   A 16x4 f32:  lane L holds M = L&15;  vgpr0 = A[M][k0 + 2*(L>>4)],
//                                        vgpr1 = A[M][k0 + 2*(L>>4) + 1]
//   B 4x16 f32:  mirrored:  .x = B[kc*4 + 2*(L>>4)][L&15],
//                           .y = B[kc*4 + 2*(L>>4)+1][L&15]
//   C/D 16x16:   vgpr r in lane L = C[r + 8*(L>>4)][L&15]
// ---------------------------------------------------------------------------

// One GRU cell step for a 16-batch tile: gates via WMMA, elementwise via LDS.
__device__ __forceinline__ void gru_step(
    const v2f Xa[9], float* sH,
    const v2f (*WihB)[32], const v2f (*WhhB)[32],
    const float* bih, const float* bhh,
    float* sGi, float* sGh, bool out_tanh,
    int lane, int m, int khalf, int rbase) {
  // h (16x20) -> A fragments, 5 K-chunks (exact, no padding)
  v2f Ha[5];
#pragma unroll
  for (int kc = 0; kc < 5; ++kc) {
    int k = kc * 4 + khalf;
    v2f a; a.x = sH[m * H_N + k]; a.y = sH[m * H_N + k + 1];
    Ha[kc] = a;
  }
  v8f gi[4], gh[4];
#pragma unroll
  for (int nt = 0; nt < 4; ++nt) {
    float bv = bih[nt * 16 + m];
    v8f acc = {bv, bv, bv, bv, bv, bv, bv, bv};     // bias broadcast over M
#pragma unroll
    for (int kc = 0; kc < 9; ++kc) acc = wmma4(Xa[kc], WihB[kc * 4 + nt][lane], acc);
    gi[nt] = acc;
    float bv2 = bhh[nt * 16 + m];
    v8f acc2 = {bv2, bv2, bv2, bv2, bv2, bv2, bv2, bv2};
#pragma unroll
    for (int kc = 0; kc < 5; ++kc) acc2 = wmma4(Ha[kc], WhhB[kc * 4 + nt][lane], acc2);
    gh[nt] = acc2;
  }
  // D-fragments -> LDS (16 x 64, cols >= 60 are zero padding)
#pragma unroll
  for (int nt = 0; nt < 4; ++nt)
#pragma unroll
    for (int r = 0; r < 8; ++r) {
      sGi[(rbase + r) * 64 + nt * 16 + m] = gi[nt][r];
      sGh[(rbase + r) * 64 + nt * 16 + m] = gh[nt][r];
    }
  __syncthreads();
  // GRU elementwise: 16*20 = 320 elements, 10 per lane
#pragma unroll
  for (int q = 0; q < 10; ++q) {
    int idx = q * 32 + lane;
    int bb = idx / H_N, j = idx % H_N;
    float ir  = sGi[bb * 64 + j],           hr = sGh[bb * 64 + j];
    float iz  = sGi[bb * 64 + H_N + j],     hz = sGh[bb * 64 + H_N + j];
    float in_ = sGi[bb * 64 + 2 * H_N + j], hn = sGh[bb * 64 + 2 * H_N + j];
    float rr = sigm(ir + hr);
    float zz = sigm(iz + hz);
    float nn = tanhf(in_ + rr * hn);
    float hp = sH[bb * H_N + j];
    float hv = (1.0f - zz) * nn + zz * hp;
    if (out_tanh) hv = tanhf(hv);             // decoder wraps gru_cell in tanh
    sH[bb * H_N + j] = hv;
  }
  __syncthreads();
}

// x = tanh(h @ Wfc.T + bfc): WMMA 20->36, write to sX (row stride 48) + global.
__device__ __forceinline__ void fc_step(
    const float* sH, const v2f (*WfcB)[32], const float* sBfc,
    float* sX, float* __restrict__ xs, int b0, int s_out,
    int lane, int m, int khalf, int rbase) {
  v2f Ha[5];
#pragma unroll
  for (int kc = 0; kc < 5; ++kc) {
    int k = kc * 4 + khalf;
    v2f a; a.x = sH[m * H_N + k]; a.y = sH[m * H_N + k + 1];
    Ha[kc] = a;
  }
  v8f xg[3];
#pragma unroll
  for (int nt = 0; nt < 3; ++nt) {
    float bv = sBfc[nt * 16 + m];
    v8f acc = {bv, bv, bv, bv, bv, bv, bv, bv};
#pragma unroll
    for (int kc = 0; kc < 5; ++kc) acc = wmma4(Ha[kc], WfcB[kc * 3 + nt][lane], acc);
    xg[nt] = acc;
  }
#pragma unroll
  for (int nt = 0; nt < 3; ++nt)
#pragma unroll
    for (int r = 0; r < 8; ++r)
      sX[(rbase + r) * 48 + nt * 16 + m] = xg[nt][r];
  __syncthreads();
  // tanh + time-flipped store: 16*36 = 576 elements, 18 per lane
#pragma unroll
  for (int q = 0; q < 18; ++q) {
    int idx = q * 32 + lane;
    int bb = idx / F_N, f = idx % F_N;
    float v = tanhf(sX[bb * 48 + f]);
    sX[bb * 48 + f] = v;
    xs[((size_t)(b0 + bb) * S_N + s_out) * F_N + f] = v;
  }
  __syncthreads();
}

// ---------------------------------------------------------------------------
// Kernel 2: full recurrence. One wave (32 threads) per 16-batch tile;
// 64 independent chains = the max parallelism the recurrence allows.
// ---------------------------------------------------------------------------
__global__ void __launch_bounds__(32)
gru_ae_kernel(const float* __restrict__ xin,
              const float* __restrict__ Wih1, const float* __restrict__ Whh1,
              const float* __restrict__ bih1, const float* __restrict__ bhh1,
              const float* __restrict__ Wih2, const float* __restrict__ Whh2,
              const float* __restrict__ bih2, const float* __restrict__ bhh2,
              const float* __restrict__ Wfc,  const float* __restrict__ bfc,
              float* __restrict__ xs) {
  // Prebuilt B-operand fragments (weights transposed into per-lane WMMA layout)
  __shared__ v2f sWihB1[36][32];   // 9 K-chunks x 4 N-tiles
  __shared__ v2f sWhhB1[20][32];   // 5 x 4
  __shared__ v2f sWihB2[36][32];
  __shared__ v2f sWhhB2[20][32];
  __shared__ v2f sWfcB[15][32];    // 5 x 3
  __shared__ float sBih1[64], sBhh1[64], sBih2[64], sBhh2[64], sBfc[48];
  __shared__ float sGi[16 * 64], sGh[16 * 64];  // gate scratch
  __shared__ float sX[16 * 48];                 // decoder x tile (padded)
  __shared__ float sH[16 * H_N];                // hidden state tile

  const int lane  = threadIdx.x;
  const int m     = lane & 15;
  const int khalf = (lane >> 4) << 1;   // 0 or 2
  const int rbase = (lane >> 4) << 3;   // 0 or 8
  const int b0    = blockIdx.x * 16;

  // ---- build weight fragments: B[k][n] = W[n][k] (zero-padded n) ----
  for (int kc = 0; kc < 9; ++kc)
    for (int nt = 0; nt < 4; ++nt) {
      int n = nt * 16 + m, k = kc * 4 + khalf;
      v2f w, w2;
      w.x  = (n < G_N) ? Wih1[n * F_N + k]     : 0.0f;
      w.y  = (n < G_N) ? Wih1[n * F_N + k + 1] : 0.0f;
      w2.x = (n < G_N) ? Wih2[n * F_N + k]     : 0.0f;
      w2.y = (n < G_N) ? Wih2[n * F_N + k + 1] : 0.0f;
      sWihB1[kc * 4 + nt][lane] = w;
      sWihB2[kc * 4 + nt][lane] = w2;
    }
  for (int kc = 0; kc < 5; ++kc)
    for (int nt = 0; nt < 4; ++nt) {
      int n = nt * 16 + m, k = kc * 4 + khalf;
      v2f w, w2;
      w.x  = (n < G_N) ? Whh1[n * H_N + k]     : 0.0f;
      w.y  = (n < G_N) ? Whh1[n * H_N + k + 1] : 0.0f;
      w2.x = (n < G_N) ? Whh2[n * H_N + k]     : 0.0f;
      w2.y = (n < G_N) ? Whh2[n * H_N + k + 1] : 0.0f;
      sWhhB1[kc * 4 + nt][lane] = w;
      sWhhB2[kc * 4 + nt][lane] = w2;
    }
  for (int kc = 0; kc < 5; ++kc)
    for (int nt = 0; nt < 3; ++nt) {
      int n = nt * 16 + m, k = kc * 4 + khalf;
      v2f w;
      w.x = (n < F_N) ? Wfc[n * H_N + k]     : 0.0f;
      w.y = (n < F_N) ? Wfc[n * H_N + k + 1] : 0.0f;
      sWfcB[kc * 3 + nt][lane] = w;
    }
  for (int i = lane; i < 64; i += 32) {
    sBih1[i] = (i < G_N) ? bih1[i] : 0.0f;
    sBhh1[i] = (i < G_N) ? bhh1[i] : 0.0f;
    sBih2[i] = (i < G_N) ? bih2[i] : 0.0f;
    sBhh2[i] = (i < G_N) ? bhh2[i] : 0.0f;
  }
  for (int i = lane; i < 48; i += 32) sBfc[i] = (i < F_N) ? bfc[i] : 0.0f;
  for (int i = lane; i < 16 * H_N; i += 32) sH[i] = 0.0f;   // h0 = 0
  __syncthreads();

  // ---- encoder: S steps, x prefetched one step ahead ----
  const float* xbase = xin + (size_t)(b0 + m) * S_N * F_N;
  v2f Xa[9], Xn[9];
#pragma unroll
  for (int kc = 0; kc < 9; ++kc)
    Xa[kc] = *(const v2f*)(xbase + kc * 4 + khalf);
#pragma unroll 1
  for (int t = 0; t < S_N; ++t) {
    if (t + 1 < S_N) {
      const float* xr = xbase + (size_t)(t + 1) * F_N;
#pragma unroll
      for (int kc = 0; kc < 9; ++kc)
        Xn[kc] = *(const v2f*)(xr + kc * 4 + khalf);
    }
    gru_step(Xa, sH, sWihB1, sWhhB1, sBih1, sBhh1, sGi, sGh,
             /*out_tanh=*/false, lane, m, khalf, rbase);
#pragma unroll
    for (int kc = 0; kc < 9; ++kc) Xa[kc] = Xn[kc];
  }

  // h = tanh(h_last)
  for (int i = lane; i < 16 * H_N; i += 32) sH[i] = tanhf(sH[i]);
  __syncthreads();

  // x0 -> output time index S-1 (output is time-flipped)
  fc_step(sH, sWfcB, sBfc, sX, xs, b0, S_N - 1, lane, m, khalf, rbase);

  // ---- autoregressive decoder ----
#pragma unroll 1
  for (int t = 1; t < S_N; ++t) {
    v2f Xd[9];
#pragma unroll
    for (int kc = 0; kc < 9; ++kc) {
      int k = kc * 4 + khalf;
      v2f a; a.x = sX[m * 48 + k]; a.y = sX[m * 48 + k + 1];
      Xd[kc] = a;
    }
    gru_step(Xd, sH, sWihB2, sWhhB2, sBih2, sBhh2, sGi, sGh,
             /*out_tanh=*/true, lane, m, khalf, rbase);
    fc_step(sH, sWfcB, sBfc, sX, xs, b0, S_N - 1 - t, lane, m, khalf, rbase);
  }
}

// ---------------------------------------------------------------------------
extern "C" void kernel_launch(void* const* d_in, const int* in_sizes, int n_in,
                              void* d_out, int out_size, void* d_ws, size_t ws_size,
                              hipStream_t stream) {
  (void)in_sizes; (void)n_in; (void)out_size; (void)d_ws; (void)ws_size;
  const float* x    = (const float*)d_in[0];
  const int*   oh   = (const int*)d_in[1];
  const float* e0   = (const float*)d_in[2];
  const float* e1   = (const float*)d_in[3];
  const float* e2   = (const float*)d_in[4];
  const float* e3   = (const float*)d_in[5];
  const float* Wih1 = (const float*)d_in[6];
  const float* Whh1 = (const float*)d_in[7];
  const float* bih1 = (const float*)d_in[8];
  const float* bhh1 = (const float*)d_in[9];
  const float* Wih2 = (const float*)d_in[10];
  const float* Whh2 = (const float*)d_in[11];
  const float* bih2 = (const float*)d_in[12];
  const float* bhh2 = (const float*)d_in[13];
  const float* Wfc  = (const float*)d_in[14];
  const float* bfc  = (const float*)d_in[15];

  float* origin = (float*)d_out;                         // output 0: (B,S,F)
  float* xs     = origin + (size_t)B_N * S_N * F_N;      // output 1: (B,S,F)

  embed_concat_kernel<<<(B_N * S_N + 255) / 256, 256, 0, stream>>>(
      x, oh, e0, e1, e2, e3, origin);

  // recurrent kernel re-reads `origin` (same stream -> ordered after kernel 1)
  gru_ae_kernel<<<B_N / 16, 32, 0, stream>>>(
      origin, Wih1, Whh1, bih1, bhh1, Wih2, Whh2, bih2, bhh2, Wfc, bfc, xs);
}